// HIN2vec_49589692400134
// MI455X (gfx1250) — compile-verified
//
#include <hip/hip_runtime.h>
#include <hip/hip_bf16.h>
#include <math.h>

// Problem constants (match reference)
#define BATCH 8192
#define EDGES 4
#define KNBR  16
#define DIM   128
#define BT    16                 // batch rows per block
#define THREADS 512              // 16 wave32 waves

typedef __attribute__((ext_vector_type(2))) float v2f;
typedef __attribute__((ext_vector_type(8))) float v8f;

__device__ __forceinline__ float sigmoidf_(float x) {
    return 1.0f / (1.0f + __expf(-x));
}

__global__ __launch_bounds__(THREADS)
void HIN2vec_fused_kernel(const int*   __restrict__ neighbors,   // [B,E,K]
                          const int*   __restrict__ end_node,    // [B]
                          const int*   __restrict__ path,        // [B]
                          const float* __restrict__ emb_start,   // [N,D]
                          const float* __restrict__ emb_end,     // [N,D]
                          const float* __restrict__ emb_path,    // [P,D]
                          const float* __restrict__ W1,          // [D,D]
                          const float* __restrict__ b1,          // [D]
                          const float* __restrict__ W2,          // [E*D,D]
                          const float* __restrict__ b2,          // [D]
                          float*       __restrict__ out)         // [B]
{
    // 32 KB: agg[64][128]; stage C reuses: s[16][128] (8KB) + W2 staging (16KB)
    __shared__ __align__(16) float lds_agg[BT * EDGES * DIM];
    // 32 KB: h viewed as [64][128] == [16][512]
    __shared__ __align__(16) float lds_h[BT * EDGES * DIM];

    const int b0   = blockIdx.x * BT;
    const int wave = threadIdx.x >> 5;            // 0..15 (wave32)
    const int lane = threadIdx.x & 31;
    const int col  = lane & 15;                   // column within 16-wide tile
    const int hiHalf = lane >> 4;                 // 0 for lanes 0-15, 1 for 16-31

    // ------------------------------------------------------------------
    // Stage A: gather neighbor embeddings + mean per (b, edge-type)
    // 64 (b,e) pairs; each wave streams 512B rows as float4 per lane.
    // ------------------------------------------------------------------
    for (int pr = wave; pr < BT * EDGES; pr += 16) {
        const int bl = pr >> 2;
        const int e  = pr & 3;
        const int* nb = neighbors + (((size_t)(b0 + bl) * EDGES + e) * KNBR);
        float4 acc = make_float4(0.f, 0.f, 0.f, 0.f);
        int idx = nb[0];
        for (int k = 0; k < KNBR; ++k) {
            const float4* row = (const float4*)(emb_start + (size_t)idx * DIM);
            if (k + 1 < KNBR) {
                idx = nb[k + 1];
                __builtin_prefetch(emb_start + (size_t)idx * DIM, 0, 0);  // global_prefetch_b8
            }
            float4 v = row[lane];
            acc.x += v.x; acc.y += v.y; acc.z += v.z; acc.w += v.w;
        }
        const float inv = 1.0f / (float)KNBR;
        float4* dst = (float4*)(lds_agg + pr * DIM) + lane;
        *dst = make_float4(acc.x * inv, acc.y * inv, acc.z * inv, acc.w * inv);
    }
    __syncthreads();

    // ------------------------------------------------------------------
    // Stage B: GEMM1  h[64,128] = agg[64,128] @ W1[128,128] + b1
    // 32 tiles of 16x16; 2 tiles per wave; V_WMMA_F32_16X16X4_F32.
    // A-frag from LDS; B-frag from global (W1 = 64KB, L2/WGP$ resident).
    // ------------------------------------------------------------------
    for (int t = wave; t < 32; t += 16) {
        const int mt = t >> 3;                    // 0..3  (M tile)
        const int nt = t & 7;                     // 0..7  (N tile)
        const int n  = nt * 16 + col;
        const int m  = mt * 16 + col;
        v8f c = {};
        #pragma unroll 4
        for (int k0 = 0; k0 < DIM; k0 += 4) {
            const int kb = k0 + (hiHalf ? 2 : 0);
            v2f a, b;
            a.x = lds_agg[m * DIM + kb];
            a.y = lds_agg[m * DIM + kb + 1];
            b.x = W1[kb * DIM + n];
            b.y = W1[(kb + 1) * DIM + n];
            c = __builtin_amdgcn_wmma_f32_16x16x4_f32(false, a, false, b,
                                                      (short)0, c, false, false);
        }
        const float bias = b1[n];
        #pragma unroll
        for (int j = 0; j < 8; ++j) {
            const int mr = mt * 16 + j + (hiHalf ? 8 : 0);
            lds_h[mr * DIM + n] = c[j] + bias;    // linear layout == h[16][512] reshape
        }
    }
    __syncthreads();

    // ------------------------------------------------------------------
    // Stage C: GEMM2  s[16,128] = h[16,512] @ W2[512,128] + b2
    // 8 n-tiles, K split across wave pairs. W2 is staged chunk-wise into
    // LDS with gfx1250 async global->LDS DMA (global_load_async_to_lds_b128
    // + s_wait_asynccnt), then all waves read B-frags from LDS.
    // ------------------------------------------------------------------
    float* lds_s = lds_agg;                       // s[16][128] (agg dead after GEMM1)
    float* stage = lds_agg + BT * DIM;            // 32 rows x 128 f32 = 16 KB staging
    for (int i = threadIdx.x; i < BT * DIM; i += THREADS)
        lds_s[i] = b2[i & (DIM - 1)];

    const int nt2   = wave & 7;                   // n tile 0..7
    const int khalf = wave >> 3;                  // 0 or 1: K half
    const int n2    = nt2 * 16 + col;
    v8f c2 = {};
    const uint32_t stage_base = (uint32_t)(size_t)stage;  // LDS byte offset (low 32 bits)

    for (int chunk = 0; chunk < 16; ++chunk) {
        // Stage rows: j = wave covers W2 row chunk*16+j (K half 0) and
        // 256+chunk*16+j (K half 1). One async instr moves one 512B row
        // (32 lanes x 16B), tracked by ASYNCcnt.
        {
            const int j = wave;                   // 0..15
            const float* g0 = W2 + ((size_t)(chunk * 16 + j)       * DIM) + lane * 4;
            const float* g1 = W2 + ((size_t)(256 + chunk * 16 + j) * DIM) + lane * 4;
            const uint32_t l0 = stage_base + (uint32_t)((j * DIM + lane * 4) * 4);
            const uint32_t l1 = l0 + (uint32_t)(16 * DIM * 4);
            asm volatile("global_load_async_to_lds_b128 %0, %1, off"
                         :: "v"(l0), "v"(g0) : "memory");
            asm volatile("global_load_async_to_lds_b128 %0, %1, off"
                         :: "v"(l1), "v"(g1) : "memory");
            asm volatile("s_wait_asynccnt 0x0" ::: "memory");
        }
        __syncthreads();                          // staged chunk visible to all waves

        #pragma unroll
        for (int kk = 0; kk < 16; kk += 4) {
            const int kloc = kk + (hiHalf ? 2 : 0);            // k within 16-row chunk
            const int kg   = khalf * 256 + chunk * 16 + kloc;  // global k (0..511)
            v2f a, b;
            a.x = lds_h[col * (EDGES * DIM) + kg];
            a.y = lds_h[col * (EDGES * DIM) + kg + 1];
            b.x = stage[(khalf * 16 + kloc)     * DIM + n2];
            b.y = stage[(khalf * 16 + kloc + 1) * DIM + n2];
            c2 = __builtin_amdgcn_wmma_f32_16x16x4_f32(false, a, false, b,
                                                       (short)0, c2, false, false);
        }
        __syncthreads();                          // all waves done before restaging
    }

    #pragma unroll
    for (int j = 0; j < 8; ++j) {
        const int mr = j + (hiHalf ? 8 : 0);
        atomicAdd(&lds_s[mr * DIM + n2], c2[j]);  // ds_add_f32: merge K halves
    }
    __syncthreads();

    // ------------------------------------------------------------------
    // Stage D: out[b] = sigmoid( sum_f s[b,f] * emb_end[end,f] * sigmoid(emb_path[p,f]) )
    // one wave-row (32 lanes) per batch row; float4 loads + wave32 shuffle reduce.
    // ------------------------------------------------------------------
    {
        const int row = threadIdx.x >> 5;         // 0..15
        const int b   = b0 + row;
        const float4 sv = ((const float4*)(lds_s + row * DIM))[lane];
        const float4 ev = ((const float4*)(emb_end  + (size_t)end_node[b] * DIM))[lane];
        const float4 pv = ((const float4*)(emb_path + (size_t)path[b]     * DIM))[lane];
        float partial = sv.x * ev.x * sigmoidf_(pv.x)
                      + sv.y * ev.y * sigmoidf_(pv.y)
                      + sv.z * ev.z * sigmoidf_(pv.z)
                      + sv.w * ev.w * sigmoidf_(pv.w);
        #pragma unroll
        for (int off = 16; off > 0; off >>= 1)
            partial += __shfl_down(partial, off, 32);
        if (lane == 0)
            out[b] = sigmoidf_(partial);
    }
}

extern "C" void kernel_launch(void* const* d_in, const int* in_sizes, int n_in,
                              void* d_out, int out_size, void* d_ws, size_t ws_size,
                              hipStream_t stream) {
    (void)in_sizes; (void)n_in; (void)out_size; (void)d_ws; (void)ws_size;
    const int*   neighbors = (const int*)  d_in[0];
    const int*   end_node  = (const int*)  d_in[1];
    const int*   path      = (const int*)  d_in[2];
    const float* emb_start = (const float*)d_in[3];
    const float* emb_end   = (const float*)d_in[4];
    const float* emb_path  = (const float*)d_in[5];
    const float* W1        = (const float*)d_in[6];
    const float* b1        = (const float*)d_in[7];
    const float* W2        = (const float*)d_in[8];
    const float* b2        = (const float*)d_in[9];
    float*       out       = (float*)d_out;

    dim3 grid(BATCH / BT);       // 512 blocks
    dim3 block(THREADS);         // 16 waves of 32
    hipLaunchKernelGGL(HIN2vec_fused_kernel, grid, block, 0, stream,
                       neighbors, end_node, path,
                       emb_start, emb_end, emb_path,
                       W1, b1, W2, b2, out);
}